// GPool_88527865905284
// MI455X (gfx1250) — compile-verified
//
#include <hip/hip_runtime.h>
#include <hip/hip_bf16.h>

typedef __attribute__((ext_vector_type(2))) float        v2f;
typedef __attribute__((ext_vector_type(8))) float        v8f;
typedef __attribute__((ext_vector_type(4))) unsigned int v4u;
typedef __attribute__((ext_vector_type(8))) int          v8i;
typedef __attribute__((ext_vector_type(4))) int          v4i;

#define C_IN   128
#define M_OUT  32
#define WAVES_PER_BLOCK 2
#define NT 8                       // 16-column tiles per wave
#define TILE_FLOATS (C_IN * 16)    // 2048 floats = 8 KB per tile

#if defined(__has_builtin)
#if __has_builtin(__builtin_amdgcn_tensor_load_to_lds) && \
    __has_builtin(__builtin_amdgcn_s_wait_tensorcnt)
#define HAVE_TDM 1
#endif
#endif

// ---------------------------------------------------------------------------
// TDM: DMA one 128x16 fp32 tile of x (row stride N) into LDS.
// D# packing per CDNA5 ISA (2-D tensor, groups 2/3 = 0).
// ---------------------------------------------------------------------------
__device__ __forceinline__ void tdm_load_tile(const float* gsrc, float* lbuf,
                                              unsigned lds_off, int N) {
#ifdef HAVE_TDM
    unsigned long long ga = (unsigned long long)(uintptr_t)gsrc;
    v4u g0;
    g0[0] = 1u;                                         // count=1, user mode
    g0[1] = lds_off;                                    // lds_addr (bytes)
    g0[2] = (unsigned)(ga & 0xFFFFFFFFu);               // global_addr[31:0]
    g0[3] = (unsigned)((ga >> 32) & 0x01FFFFFFu)        // global_addr[56:32]
          | 0x80000000u;                                // type=2 ("image")
    unsigned nd = (unsigned)N;
    v8i g1;
    g1[0] = (int)(2u << 16);                            // data_size = 4 bytes
    g1[1] = (int)((nd & 0xFFFFu) << 16);                // tensor_dim0[15:0]
    g1[2] = (int)((nd >> 16) | (128u << 16));           // tensor_dim0[31:16] | tensor_dim1
    g1[3] = (int)(16u << 16);                           // tile_dim0 = 16
    g1[4] = 128;                                        // tile_dim1 = 128
    g1[5] = (int)nd;                                    // tensor_dim0_stride[31:0] = N
    g1[6] = 0;                                          // stride0[47:32] | stride1[15:0]
    g1[7] = 0;                                          // stride1[47:16]
    v4i gz = {0, 0, 0, 0};
#if __clang_major__ >= 23
    v8i gz8 = {0, 0, 0, 0, 0, 0, 0, 0};
    __builtin_amdgcn_tensor_load_to_lds(g0, g1, gz, gz, gz8, 0);
#else
    __builtin_amdgcn_tensor_load_to_lds(g0, g1, gz, gz, 0);
#endif
#else
    // Fallback: per-lane b128 copy (wave-private buffer; LDS ops are in-order)
    const int lane = threadIdx.x & 31;
    for (int i = lane; i < C_IN * 4; i += 32) {
        int r = i >> 2, q = i & 3;
        float4 v = *(const float4*)(gsrc + (size_t)r * N + q * 4);
        *(float4*)(lbuf + r * 16 + q * 4) = v;
    }
#endif
}

__device__ __forceinline__ void tdm_wait(int outstanding) {
#ifdef HAVE_TDM
    if (outstanding)
        __builtin_amdgcn_s_wait_tensorcnt(1);
    else
        __builtin_amdgcn_s_wait_tensorcnt(0);
#endif
}

// ---------------------------------------------------------------------------
// Phase 1: score = sigmoid(W2 . relu(W1 @ x + b1))
// V_WMMA_F32_16X16X4_F32; x tiles staged into LDS by the Tensor Data Mover,
// double-buffered so the DMA of tile t+1 overlaps the WMMA chain on tile t.
// ---------------------------------------------------------------------------
__global__ void __launch_bounds__(WAVES_PER_BLOCK * 32)
gpool_score_kernel(const float* __restrict__ x,
                   const float* __restrict__ W1,
                   const float* __restrict__ b1,
                   const float* __restrict__ W2,
                   float* __restrict__ score,
                   int B, int N) {
    __shared__ float sW1[M_OUT * C_IN];                        // 16 KB
    __shared__ float sB1[M_OUT];
    __shared__ float sW2[M_OUT];
    __shared__ float xbuf[WAVES_PER_BLOCK * 2 * TILE_FLOATS];  // 32 KB

    const int tid = threadIdx.x;
    for (int i = tid; i < M_OUT * C_IN; i += blockDim.x) sW1[i] = W1[i];
    if (tid < M_OUT) { sB1[tid] = b1[tid]; sW2[tid] = W2[tid]; }
    __syncthreads();

    const int wid  = tid >> 5;
    const int lane = tid & 31;
    const int gwave = (int)blockIdx.x * WAVES_PER_BLOCK + wid;

    const int wavesPerBatch = (N >> 4) / NT;                   // 256
    const int b      = gwave / wavesPerBatch;
    const int n_base = (gwave % wavesPerBatch) * (16 * NT);
    const float* xb  = x + (size_t)b * C_IN * (size_t)N;

    float* buf0 = &xbuf[wid * 2 * TILE_FLOATS];
    float* buf1 = buf0 + TILE_FLOATS;
    // Flat LDS aperture keeps the byte offset in addr[31:0]
    const unsigned off0 = (unsigned)(uintptr_t)buf0;
    const unsigned off1 = (unsigned)(uintptr_t)buf1;

    const int col   = lane & 15;          // column in tile / M row in A frag
    const int khalf = (lane >> 4) << 1;   // 0 (lanes 0-15) or 2 (lanes 16-31)
    const int row0  = (lane >> 4) << 3;   // 0 or 8 in C/D layout

    tdm_load_tile(xb + n_base, buf0, off0, N);                 // prologue

    for (int t = 0; t < NT; ++t) {
        if (t + 1 < NT) {
            float*   nb  = ((t + 1) & 1) ? buf1 : buf0;
            unsigned no  = ((t + 1) & 1) ? off1 : off0;
            tdm_load_tile(xb + n_base + (t + 1) * 16, nb, no, N);
            tdm_wait(1);                  // tile t has landed
        } else {
            tdm_wait(0);
        }

        const float* buf = (t & 1) ? buf1 : buf0;              // row-major 128x16
        const int n0 = n_base + t * 16;

        v8f acc0 = {};                    // h rows 0..15
        v8f acc1 = {};                    // h rows 16..31
        for (int kk = 0; kk < C_IN; kk += 4) {
            v2f bf;                       // B frag (4x16) from LDS tile
            bf.x = buf[(kk + khalf)     * 16 + col];
            bf.y = buf[(kk + khalf + 1) * 16 + col];
            v2f a0, a1;                   // A frags (16x4) from LDS W1
            a0.x = sW1[col * C_IN + kk + khalf];
            a0.y = sW1[col * C_IN + kk + khalf + 1];
            a1.x = sW1[(col + 16) * C_IN + kk + khalf];
            a1.y = sW1[(col + 16) * C_IN + kk + khalf + 1];

            acc0 = __builtin_amdgcn_wmma_f32_16x16x4_f32(false, a0, false, bf,
                                                         (short)0, acc0, false, false);
            acc1 = __builtin_amdgcn_wmma_f32_16x16x4_f32(false, a1, false, bf,
                                                         (short)0, acc1, false, false);
        }

        float s = 0.0f;
#pragma unroll
        for (int v = 0; v < 8; ++v) {
            int r0 = row0 + v;
            float h0 = acc0[v] + sB1[r0];
            h0 = h0 > 0.0f ? h0 : 0.0f;
            s += sW2[r0] * h0;
            int r1 = 16 + row0 + v;
            float h1 = acc1[v] + sB1[r1];
            h1 = h1 > 0.0f ? h1 : 0.0f;
            s += sW2[r1] * h1;
        }
        s += __shfl_xor(s, 16, 32);       // combine the two lane halves

        if (lane < 16) {
            float sig = 1.0f / (1.0f + expf(-s));
            score[(size_t)b * N + n0 + col] = sig;
        }
    }
}

// ---------------------------------------------------------------------------
// Phase 2: exact top-k via full bitonic sort in LDS (32768 x u64 = 256 KB,
// fits the 320 KB CDNA5 WGP LDS). One workgroup per batch.
// Key = orderable(score bits) << 32 | (~index): descending sort reproduces
// top_k's value-desc / index-asc tie-break order exactly.
// ---------------------------------------------------------------------------
__global__ void gpool_topk_kernel(const float* __restrict__ score,
                                  int* __restrict__ topidx,
                                  float* __restrict__ topscore,
                                  int N, int K) {
    extern __shared__ unsigned long long keys[];   // N entries
    const int b   = blockIdx.x;
    const int tid = threadIdx.x;
    const int nt  = blockDim.x;
    const float* sb = score + (size_t)b * N;

    for (int i = tid; i < N; i += nt) {
        unsigned int bits = __float_as_uint(sb[i]);
        bits = (bits & 0x80000000u) ? ~bits : (bits | 0x80000000u);   // total order
        keys[i] = ((unsigned long long)bits << 32) |
                  (unsigned long long)(0xFFFFFFFFu - (unsigned int)i);
    }
    __syncthreads();

    for (int size = 2; size <= N; size <<= 1) {
        for (int stride = size >> 1; stride > 0; stride >>= 1) {
            for (int i = tid; i < (N >> 1); i += nt) {
                int a  = ((i / stride) * (stride << 1)) + (i % stride);
                int p  = a + stride;
                bool desc = ((a & size) == 0);     // descending overall
                unsigned long long ka = keys[a];
                unsigned long long kp = keys[p];
                if ((ka < kp) == desc) { keys[a] = kp; keys[p] = ka; }
            }
            __syncthreads();
        }
    }

    for (int j = tid; j < K; j += nt) {
        unsigned long long kv = keys[j];
        unsigned int bits = (unsigned int)(kv >> 32);
        bits = (bits & 0x80000000u) ? (bits & 0x7FFFFFFFu) : ~bits;   // undo mapping
        int idx = (int)(0xFFFFFFFFu - (unsigned int)(kv & 0xFFFFFFFFu));
        topidx[(size_t)b * K + j]   = idx;
        topscore[(size_t)b * K + j] = __uint_as_float(bits);
    }
}

// ---------------------------------------------------------------------------
// Phase 3: gathers. One thread per (b, j); writes coalesced in j.
// Output layout: pos_k[B,3,K] | local_feat_k[B,64,K] | feat_k[B,128,K]
// ---------------------------------------------------------------------------
__global__ void gpool_gather_kernel(const float* __restrict__ pos,
                                    const float* __restrict__ feat,
                                    const float* __restrict__ x,
                                    const int* __restrict__ topidx,
                                    const float* __restrict__ topscore,
                                    float* __restrict__ out,
                                    int B, int N, int K, int Cf, int C) {
    int t = blockIdx.x * blockDim.x + threadIdx.x;
    if (t >= B * K) return;
    const int b = t / K;
    const int j = t - b * K;

    const int   idx = topidx[(size_t)b * K + j];
    const float sc  = topscore[(size_t)b * K + j];

    float* pos_out = out;
    float* lf_out  = out + (size_t)B * 3 * K;
    float* fk_out  = lf_out + (size_t)B * Cf * K;

#pragma unroll
    for (int d = 0; d < 3; ++d)
        pos_out[((size_t)b * 3 + d) * K + j] = pos[((size_t)b * 3 + d) * N + idx];
    for (int c = 0; c < Cf; ++c)
        lf_out[((size_t)b * Cf + c) * K + j] = feat[((size_t)b * Cf + c) * N + idx];
    for (int c = 0; c < C; ++c)
        fk_out[((size_t)b * C + c) * K + j] = x[((size_t)b * C + c) * N + idx] * sc;
}

// ---------------------------------------------------------------------------
extern "C" void kernel_launch(void* const* d_in, const int* in_sizes, int n_in,
                              void* d_out, int out_size, void* d_ws, size_t ws_size,
                              hipStream_t stream) {
    const float* pos  = (const float*)d_in[0];   // (16, 3, 32768)
    const float* feat = (const float*)d_in[1];   // (16, 64, 32768)
    const float* x    = (const float*)d_in[2];   // (16, 128, 32768)
    const float* W1   = (const float*)d_in[3];   // (32, 128)
    const float* b1   = (const float*)d_in[4];   // (32,)
    const float* W2   = (const float*)d_in[5];   // (32,)

    const int B = 16, C = 128, Cf = 64;
    const int N = in_sizes[2] / (B * C);         // 32768
    const int K = N / 4;                         // 8192

    // workspace: score[B*N] | topidx[B*K] | topscore[B*K]
    float* score    = (float*)d_ws;
    int*   topidx   = (int*)((char*)d_ws + (size_t)B * N * sizeof(float));
    float* topscore = (float*)((char*)d_ws + (size_t)B * N * sizeof(float)
                                           + (size_t)B * K * sizeof(int));

    // Phase 1: each wave covers NT*16 = 128 columns; 2 waves (64 thr) / block
    {
        int totalWaves = B * ((N >> 4) / NT);    // 4096
        dim3 blk(WAVES_PER_BLOCK * 32);
        dim3 grd((unsigned)(totalWaves / WAVES_PER_BLOCK));
        gpool_score_kernel<<<grd, blk, 0, stream>>>(x, W1, b1, W2, score, B, N);
    }

    // Phase 2: one workgroup per batch, 256 KB dynamic LDS bitonic sort
    {
        size_t ldsBytes = (size_t)N * sizeof(unsigned long long);
        hipFuncSetAttribute((const void*)gpool_topk_kernel,
                            hipFuncAttributeMaxDynamicSharedMemorySize,
                            (int)ldsBytes);
        gpool_topk_kernel<<<dim3((unsigned)B), dim3(1024), ldsBytes, stream>>>(
            score, topidx, topscore, N, K);
    }

    // Phase 3: gathers
    {
        int total = B * K;
        gpool_gather_kernel<<<dim3((unsigned)((total + 255) / 256)), dim3(256), 0, stream>>>(
            pos, feat, x, topidx, topscore, (float*)d_out, B, N, K, Cf, C);
    }
}